// Apply2DTform_5506148074183
// MI455X (gfx1250) — compile-verified
//
#include <hip/hip_runtime.h>

// Spatial-transformer bilinear sampling, MI455X (gfx1250).
//
// Memory-bound op: ~64MB HBM traffic, ~0.34 GFLOP -> HBM roofline ~2.9us.
// WMMA is intentionally NOT used: there is no matrix contraction here, only
// data-dependent gathers (L2-resident, 192MB L2 holds the full 32MB image set).
// CDNA5-specific paths used: global_load_async_to_lds_b32 + s_wait_asynccnt
// (ASYNCcnt) to broadcast the per-batch affine params into LDS, ds_load reads,
// and non-temporal float4 stores for the write stream.

#define OUT_H 1024
#define OUT_W 1024
#define IMG_H 1024
#define IMG_W 1024

typedef __attribute__((ext_vector_type(4))) float v4f;

__device__ __forceinline__ float tap(const float* __restrict__ imgb, int xi, int yi) {
    // Reference pads the image with one zero row/col at index 1024; indices are
    // pre-clamped to [0,1024], so index==1024 reads the zero pad.
    return (xi < IMG_H && yi < IMG_W) ? imgb[xi * IMG_W + yi] : 0.0f;
}

__global__ __launch_bounds__(256) void stn_bilinear_kernel(
    const float* __restrict__ Img,    // (B, 1024, 1024, 1) fp32
    const float* __restrict__ Tform,  // (B, 6) fp32
    float* __restrict__ Out)          // (B, 1024, 1024, 1) fp32
{
    __shared__ float sT[8];

    const int tid = threadIdx.x;          // 0..255  (8 wave32 per block)
    const int b   = blockIdx.z;

    // --- CDNA5 async path: stage the 6 uniform params into LDS -------------
    if (tid < 6) {
        unsigned lds_off = (unsigned)(unsigned long long)(&sT[tid]); // low 32b of
                                                                     // generic ptr
                                                                     // == LDS addr
        unsigned long long gaddr = (unsigned long long)(Tform + b * 6 + tid);
        asm volatile("global_load_async_to_lds_b32 %0, %1, off"
                     :: "v"(lds_off), "v"(gaddr)
                     : "memory");
    }
    asm volatile("s_wait_asynccnt 0x0" ::: "memory");
    __syncthreads();

    const float m00 = sT[0];
    const float m01 = sT[1];
    const float m10 = sT[2];
    const float m11 = sT[3];
    const float v0  = sT[4];
    const float v1  = sT[5];

    // --- 32x32 output tile; thread handles 4 consecutive j (float4 store) --
    const int i  = blockIdx.y * 32 + (tid >> 3);          // output row  (ax dim)
    const int j0 = blockIdx.x * 32 + (tid & 7) * 4;       // output col  (ay dim)

    const float step = 2.0f / 1023.0f;                    // linspace(-1,1,1024)
    const float axi  = fmaf(step, (float)i, -1.0f);

    // x_s = m00*ax + m01*ay + v0 ; y_s = m10*ax + m11*ay + v1
    const float base_x = fmaf(m00, axi, v0);
    const float base_y = fmaf(m10, axi, v1);

    const float* __restrict__ imgb = Img + (size_t)b * (size_t)(IMG_H * IMG_W);

    float out4[4];
#pragma unroll
    for (int k = 0; k < 4; ++k) {
        const int   j  = j0 + k;
        const float ay = fmaf(step, (float)j, -1.0f);

        const float xs = fmaf(m01, ay, base_x);
        const float ys = fmaf(m11, ay, base_y);

        // x = 0.5*(xs+1)*1023 ; y = 0.5*(ys+1)*1023
        const float x = (xs + 1.0f) * 511.5f;
        const float y = (ys + 1.0f) * 511.5f;

        const float fx = floorf(x);
        const float fy = floorf(y);

        // clip(x0,0,1024), clip(x0+1,0,1024) — done in float (values integral)
        const float x0f = fminf(fmaxf(fx,        0.0f), 1024.0f);
        const float x1f = fminf(fmaxf(fx + 1.0f, 0.0f), 1024.0f);
        const float y0f = fminf(fmaxf(fy,        0.0f), 1024.0f);
        const float y1f = fminf(fmaxf(fy + 1.0f, 0.0f), 1024.0f);

        const int x0 = (int)x0f;
        const int x1 = (int)x1f;
        const int y0 = (int)y0f;
        const int y1 = (int)y1f;

        const float I00 = tap(imgb, x0, y0);
        const float I01 = tap(imgb, x0, y1);
        const float I10 = tap(imgb, x1, y0);
        const float I11 = tap(imgb, x1, y1);

        const float wx1 = x1f - x;   // weight toward x0 row
        const float wx0 = x - x0f;   // weight toward x1 row
        const float wy1 = y1f - y;
        const float wy0 = y - y0f;

        float acc = (wx1 * wy1) * I00;
        acc = fmaf(wx1 * wy0, I01, acc);
        acc = fmaf(wx0 * wy1, I10, acc);
        acc = fmaf(wx0 * wy0, I11, acc);
        out4[k] = acc;
    }

    v4f res = { out4[0], out4[1], out4[2], out4[3] };
    v4f* dst = (v4f*)(Out + (size_t)b * (size_t)(OUT_H * OUT_W)
                          + (size_t)i * OUT_W + j0);
    __builtin_nontemporal_store(res, dst);   // keep write stream out of L2's way
}

extern "C" void kernel_launch(void* const* d_in, const int* in_sizes, int n_in,
                              void* d_out, int out_size, void* d_ws, size_t ws_size,
                              hipStream_t stream) {
    const float* Img   = (const float*)d_in[0];   // (B,1024,1024,1) fp32
    const float* Tform = (const float*)d_in[1];   // (B,6) fp32
    float*       Out   = (float*)d_out;           // (B,1024,1024,1) fp32

    const int B = in_sizes[1] / 6;                // 8

    dim3 grid(OUT_W / 32, OUT_H / 32, B);         // 32 x 32 x B blocks
    dim3 block(256);                              // 8 wave32 per block
    stn_bilinear_kernel<<<grid, block, 0, stream>>>(Img, Tform, Out);
}